// MFDWCFeatureExtractor_10033043603887
// MI455X (gfx1250) — compile-verified
//
#include <hip/hip_runtime.h>
#include <math.h>

// ---------------------------------------------------------------------------
// MFDWC feature extractor for MI455X (gfx1250).
// - Waveform span staged to LDS via GLOBAL_LOAD_ASYNC_TO_LDS_B32 (ASYNCcnt),
//   overlapped with cos/sin table construction.
// - DFT-as-GEMM via V_WMMA_F32_16X16X4_F32 (f32 in, f32 acc), fused preemph/
//   window/framing in LDS, fused mel projection + log.
// - Per-batch DWT/delta/mean/std with f64 accumulation.
// ---------------------------------------------------------------------------

typedef float v2f __attribute__((ext_vector_type(2)));
typedef float v8f __attribute__((ext_vector_type(8)));

#define SRLEN     441000
#define NBATCH    16
#define FRAME_LEN 882
#define FRAME_HOP 441
#define NFRAMES   999
#define NROWS     (NBATCH * NFRAMES)   // 15984 = 999 * 16 (multiple of 16)
#define KPAD      896                  // frame length padded (zeros beyond 882)
#define NBINS     1025
#define BINPAD    1040                 // 65 * 16
#define NKTILES   65
#define NMEL      60
#define NFEAT     90
#define WAVBUF    7940                 // max contiguous samples a tile needs

// smem partition (floats): [sCos 2048][sSin 2048][sF 16*896][X max(sPow,sWav)]
#define OFF_COS   0
#define OFF_SIN   2048
#define OFF_F     4096
#define OFF_X     (4096 + 16 * KPAD)               // 18432
#define SMEM_F32  (OFF_X + 16 * BINPAD)            // 35072 floats = 140288 B

#define TWO_PI_OVER_2048 3.0679615757712823e-03f
#define TWO_PI_OVER_881  7.1330141181512709e-03f

__launch_bounds__(256)
__global__ void dft_mel_kernel(const float* __restrict__ wav,
                               const float* __restrict__ mf,
                               float* __restrict__ melbuf)
{
    __shared__ float smem[SMEM_F32];
    float* sCos = smem + OFF_COS;
    float* sSin = smem + OFF_SIN;
    float* sF   = smem + OFF_F;
    float* sX   = smem + OFF_X;        // sWav during staging, sPow afterwards

    const int tid  = threadIdx.x;
    const int row0 = blockIdx.x * 16;  // 999 blocks, 16 frames each

    // --- contiguous flat sample span covering all 16 frames (incl. preemph) ---
    const int b0 = row0 / NFRAMES,        t0 = row0 - b0 * NFRAMES;
    const int b1 = (row0 + 15) / NFRAMES, t1 = (row0 + 15) - b1 * NFRAMES;
    int fstart = b0 * SRLEN + t0 * FRAME_HOP - 1;      // one extra for x[s-1]
    if (fstart < 0) fstart = 0;
    const int fend  = b1 * SRLEN + t1 * FRAME_HOP + FRAME_LEN - 1;
    const int count = fend - fstart + 1;               // <= WAVBUF

    // --- async DMA global -> LDS (bypasses VGPRs, tracked by ASYNCcnt) ---
    {
        const float* base = wav + fstart;
        for (int i = tid; i < count; i += 256) {
            unsigned ldsAddr = (unsigned)(size_t)(sX + i);   // flat[31:0] = LDS offset
            unsigned gOff    = (unsigned)i * 4u;
            asm volatile("global_load_async_to_lds_b32 %0, %1, %2"
                         :: "v"(ldsAddr), "v"(gOff), "s"(base) : "memory");
        }
    }

    // --- overlap: cos/sin tables (no global deps) ---
    for (int p = tid; p < 2048; p += 256) {
        float ang = (float)p * TWO_PI_OVER_2048;
        sCos[p] = cosf(ang);
        sSin[p] = sinf(ang);
    }

    asm volatile("s_wait_asynccnt 0x0" ::: "memory");
    __syncthreads();

    // --- fused pre-emphasis + Hamming window + framing (LDS -> LDS) ---
    for (int i = tid; i < 16 * KPAD; i += 256) {
        int r = i / KPAD;
        int n = i - r * KPAD;
        int row = row0 + r;
        int b = row / NFRAMES;
        int t = row - b * NFRAMES;
        float v = 0.0f;
        if (n < FRAME_LEN) {
            int s    = t * FRAME_HOP + n;          // per-batch sample index
            int flat = b * SRLEN + s - fstart;     // LDS index into sX
            float e = sX[flat] - ((s > 0) ? 0.97f * sX[flat - 1] : 0.0f);
            float w = 0.54f - 0.46f * cosf(TWO_PI_OVER_881 * (float)n);
            v = e * w;
        }
        sF[i] = v;
    }
    __syncthreads();                               // sX (sWav) dead; becomes sPow

    const int lane = tid & 31;
    const int wave = tid >> 5;
    const int m16  = lane & 15;
    const int hi   = lane >> 4;                    // half-wave K split

    // ------------------ Phase 1: DFT power spectrum ------------------
    // D(16 frames x 16 bins) += A(16x4 frames x n) * B(4x16 n x bins)
    for (int kt = wave; kt < NKTILES; kt += 8) {
        const int k = kt * 16 + m16;               // this lane's bin (N)
        v8f accR = {0.f,0.f,0.f,0.f,0.f,0.f,0.f,0.f};
        v8f accI = {0.f,0.f,0.f,0.f,0.f,0.f,0.f,0.f};
        int pbase = (2 * k * hi) & 2047;           // phase at n = 2*hi
        const int pstep = (4 * k) & 2047;          // phase advance per 4-K step
        const float* fA = sF + m16 * KPAD + 2 * hi;

        for (int n0 = 0; n0 < 884; n0 += 4) {      // 221 steps, cols >=882 are 0
            v2f a; a.x = fA[n0]; a.y = fA[n0 + 1];
            int p1 = (pbase + k) & 2047;
            v2f bc; bc.x = sCos[pbase]; bc.y = sCos[p1];
            v2f bs; bs.x = sSin[pbase]; bs.y = sSin[p1];
            accR = __builtin_amdgcn_wmma_f32_16x16x4_f32(
                       false, a, false, bc, (short)0, accR, false, false);
            accI = __builtin_amdgcn_wmma_f32_16x16x4_f32(
                       false, a, false, bs, (short)0, accI, false, false);
            pbase = (pbase + pstep) & 2047;
        }
        // D layout: VGPR j -> row j + 8*hi, col = kt*16 + m16
#pragma unroll
        for (int j = 0; j < 8; ++j) {
            int fr = j + 8 * hi;
            sX[fr * BINPAD + kt * 16 + m16] = accR[j] * accR[j] + accI[j] * accI[j];
        }
    }
    __syncthreads();

    // ------------------ Phase 2: mel projection + log ------------------
    // D(16 frames x 16 mels) += A(16x4 frames x bin) * B(4x16 bin x mel)
    if (wave < 4) {
        const int m0 = wave * 16;
        const int mm = m0 + m16;                   // mel index (N)
        if (mm < NMEL) __builtin_prefetch(mf + mm * NBINS, 0, 3);
        v8f acc = {0.f,0.f,0.f,0.f,0.f,0.f,0.f,0.f};
        const float* aP = sX + m16 * BINPAD + 2 * hi;

        for (int k0 = 0; k0 < 1028; k0 += 4) {     // 257 steps, K = 0..1027
            v2f a; a.x = aP[k0]; a.y = aP[k0 + 1];
            int kb = k0 + 2 * hi;
            v2f b;
            b.x = (mm < NMEL && kb     < NBINS) ? mf[mm * NBINS + kb]     : 0.0f;
            b.y = (mm < NMEL && kb + 1 < NBINS) ? mf[mm * NBINS + kb + 1] : 0.0f;
            acc = __builtin_amdgcn_wmma_f32_16x16x4_f32(
                      false, a, false, b, (short)0, acc, false, false);
        }
#pragma unroll
        for (int j = 0; j < 8; ++j) {
            int fr = j + 8 * hi;
            if (mm < NMEL)
                melbuf[(row0 + fr) * NMEL + mm] = logf(acc[j] + 1e-10f);
        }
    }
}

// ------------------ per-batch DWT + delta + mean/std ------------------
__launch_bounds__(96)
__global__ void stats_kernel(const float* __restrict__ melbuf,
                             float* __restrict__ out)
{
    const int b = blockIdx.x;
    const int j = threadIdx.x;
    if (j >= NFEAT) return;

    const float inv = 0.70710678118654752f;  // 1/sqrt(2)
    const float* L = melbuf + b * NFRAMES * NMEL;
    const int i    = j % 30;
    const int kind = j / 30;                 // 0: cA, 1: delta, 2: cD

    double s = 0.0, s2 = 0.0;
    for (int t = 0; t < NFRAMES; ++t) {
        float f;
        if (kind == 0) {
            f = (L[t * NMEL + 2 * i] + L[t * NMEL + 2 * i + 1]) * inv;
        } else if (kind == 1) {
            int tp = (t + 1 < NFRAMES) ? t + 1 : NFRAMES - 1;
            int tm = (t - 1 >= 0) ? t - 1 : 0;
            float cAp = (L[tp * NMEL + 2 * i] + L[tp * NMEL + 2 * i + 1]) * inv;
            float cAm = (L[tm * NMEL + 2 * i] + L[tm * NMEL + 2 * i + 1]) * inv;
            f = cAp - cAm;
        } else {
            f = (L[t * NMEL + 2 * i] - L[t * NMEL + 2 * i + 1]) * inv;
        }
        s  += (double)f;
        s2 += (double)f * (double)f;
    }
    double mean = s / (double)NFRAMES;
    double var  = (s2 - s * s / (double)NFRAMES) / (double)(NFRAMES - 1);
    if (var < 0.0) var = 0.0;
    out[b * 2 * NFEAT + j]         = (float)mean;
    out[b * 2 * NFEAT + NFEAT + j] = (float)sqrt(var);
}

extern "C" void kernel_launch(void* const* d_in, const int* in_sizes, int n_in,
                              void* d_out, int out_size, void* d_ws, size_t ws_size,
                              hipStream_t stream)
{
    const float* wav = (const float*)d_in[0];   // (16, 441000) f32
    const float* mf  = (const float*)d_in[1];   // (60, 1025)  f32
    float* out    = (float*)d_out;              // (16, 180)   f32
    float* melbuf = (float*)d_ws;               // 15984*60 f32 = 3.84 MB

    dft_mel_kernel<<<NROWS / 16, 256, 0, stream>>>(wav, mf, melbuf);
    stats_kernel<<<NBATCH, 96, 0, stream>>>(melbuf, out);
}